// RMABSolver_12257836662954
// MI455X (gfx1250) — compile-verified
//
#include <hip/hip_runtime.h>
#include <hip/hip_bf16.h>
#include <math.h>

// RMAB solver: Whittle indices (Cramer 3x3 solves) + Sinkhorn soft-top-k.
// One workgroup per batch row; C/f in registers; g-reduction via wave32
// shuffles + V_WMMA_F32_16X16X4_F32 lane-sum + tiny LDS cross-wave combine.

typedef __attribute__((ext_vector_type(2))) float v2f;
typedef __attribute__((ext_vector_type(8))) float v8f;

#define GAMMA_D  0.99f
#define EPS_V    0.01f
#define N_ITER   200
#define N_ELEM   2048
#define K_BUDGET 40
#define TPB      256
#define EPT      (N_ELEM / TPB)   // 8 elements per thread
#define NWAVES   (TPB / 32)       // 8 waves per workgroup

__device__ __forceinline__ float det3(float a00, float a01, float a02,
                                      float a10, float a11, float a12,
                                      float a20, float a21, float a22) {
  return a00 * (a11 * a22 - a12 * a21)
       - a01 * (a10 * a22 - a12 * a20)
       + a02 * (a10 * a21 - a11 * a20);
}

// Solve 3x3 system A c = b via Cramer's rule.
__device__ __forceinline__ void solve3(float a00, float a01, float a02,
                                       float a10, float a11, float a12,
                                       float a20, float a21, float a22,
                                       float b0, float b1, float b2,
                                       float& c0, float& c1, float& c2) {
  float inv = 1.0f / det3(a00, a01, a02, a10, a11, a12, a20, a21, a22);
  c0 = det3(b0, a01, a02, b1, a11, a12, b2, a21, a22) * inv;
  c1 = det3(a00, b0, a02, a10, b1, a12, a20, b2, a22) * inv;
  c2 = det3(a00, a01, b0, a10, a11, b1, a20, a21, b2) * inv;
}

__global__ __launch_bounds__(TPB) void rmab_sinkhorn_kernel(
    const float* __restrict__ T, const int* __restrict__ state,
    float* __restrict__ out) {
  const int b    = blockIdx.x;
  const int tid  = threadIdx.x;
  const int lane = tid & 31;
  const int wid  = tid >> 5;

  __shared__ float redMax0[NWAVES];
  __shared__ float redMax1[NWAVES];
  __shared__ float redSum0[NWAVES];
  __shared__ float redSum1[NWAVES];

  float C0[EPT], C1[EPT], F[EPT];

  // Constant selector A-matrix for WMMA lane reduction:
  //   row 0 = [1,0,1,0]  -> D(0,n) = B(0,n)+B(2,n)  (column-0 partials)
  //   row 1 = [0,1,0,1]  -> D(1,n) = B(1,n)+B(3,n)  (column-1 partials)
  // A layout (16x4 f32): lane l<16 holds {K0,K1} of row l; lane l>=16 holds
  // {K2,K3} of row l-16.
  v2f amat;
  amat.x = (lane == 0 || lane == 16) ? 1.0f : 0.0f;
  amat.y = (lane == 1 || lane == 17) ? 1.0f : 0.0f;

  // ---------------- Phase 1: Whittle indices + raw cost matrix ----------------
  float localCmax = 0.0f;  // C_raw >= 0
  for (int k = 0; k < EPT; ++k) {
    const int i = tid + k * TPB;
    const float* tp = T + ((size_t)b * N_ELEM + i) * 8;
    float4 lo = *(const float4*)tp;        // t000 t001 t010 t011
    float4 hi = *(const float4*)(tp + 4);  // t100 t101 t110 t111
    const float t000 = lo.x, t001 = lo.y, t010 = lo.z, t011 = lo.w;
    const float t100 = hi.x, t101 = hi.y, t110 = hi.z, t111 = hi.w;
    const float g = GAMMA_D;

    // rows r1..r4
    const float r10 = 1.0f, r11 = g * t000 - 1.0f, r12 = g * t001;
    const float r20 = 0.0f, r21 = g * t010 - 1.0f, r22 = g * t011;
    const float r30 = 1.0f, r31 = g * t100,        r32 = g * t101 - 1.0f;
    const float r40 = 0.0f, r41 = g * t110,        r42 = g * t111 - 1.0f;

    // state 0: A1=[r1;r2;r3], A2=[r1;r2;r4], b=[0,0,-1]
    float c10, c11, c12, c20, tmp1, tmp2;
    solve3(r10, r11, r12, r20, r21, r22, r30, r31, r32,
           0.0f, 0.0f, -1.0f, c10, c11, c12);
    solve3(r10, r11, r12, r20, r21, r22, r40, r41, r42,
           0.0f, 0.0f, -1.0f, c20, tmp1, tmp2);
    const bool m0 = (c10 + 1.0f + g * (t100 * c11 + t101 * c12))
                 >= (1.0f + g * (t110 * c11 + t111 * c12));
    const float w0 = m0 ? c10 : c20;

    // state 1: A1=[r3;r4;r1], A2=[r3;r4;r2], b=[-1,-1,0]
    float d10, d11, d12, d20;
    solve3(r30, r31, r32, r40, r41, r42, r10, r11, r12,
           -1.0f, -1.0f, 0.0f, d10, d11, d12);
    solve3(r30, r31, r32, r40, r41, r42, r20, r21, r22,
           -1.0f, -1.0f, 0.0f, d20, tmp1, tmp2);
    const bool m1 = (d10 + g * (t000 * d11 + t001 * d12))
                 >= (g * (t010 * d11 + t011 * d12));
    const float w1 = m1 ? d10 : d20;

    const int st = state[(size_t)b * N_ELEM + i];
    const float s = -((st == 0) ? w0 : w1);  // scores = -W_state
    const float cr0 = s * s;
    const float sm1 = s - 1.0f;
    const float cr1 = sm1 * sm1;
    C0[k] = cr0;
    C1[k] = cr1;
    localCmax = fmaxf(localCmax, fmaxf(cr0, cr1));
  }

  // Row-wide max of C_raw (normalization)
  float cm = localCmax;
  for (int off = 16; off; off >>= 1) cm = fmaxf(cm, __shfl_xor(cm, off, 32));
  if (lane == 0) redMax0[wid] = cm;
  __syncthreads();
  float Cmax = redMax0[0];
  for (int w = 1; w < NWAVES; ++w) Cmax = fmaxf(Cmax, redMax0[w]);
  __syncthreads();
  const float invCmax = 1.0f / Cmax;
  for (int k = 0; k < EPT; ++k) {
    C0[k] *= invCmax;
    C1[k] *= invCmax;
  }

  // ---------------- Phase 2: Sinkhorn iterations ----------------
  const float inv_eps     = 1.0f / EPS_V;
  const float eps_log_mu  = EPS_V * __logf(1.0f / (float)N_ELEM);
  const float eps_log_nu0 = EPS_V * __logf((float)K_BUDGET / (float)N_ELEM);
  const float eps_log_nu1 = EPS_V * __logf((float)(N_ELEM - K_BUDGET) / (float)N_ELEM);

  float g0 = 0.0f, g1 = 0.0f;

  for (int it = 0; it < N_ITER; ++it) {
    // f-update (elementwise 2-way stable logsumexp) + per-lane max of x
    float xmax0 = -3.402823466e38f, xmax1 = -3.402823466e38f;
    for (int k = 0; k < EPT; ++k) {
      const float a0 = (g0 - C0[k]) * inv_eps;
      const float a1 = (g1 - C1[k]) * inv_eps;
      const float mm = fmaxf(a0, a1);
      const float lse = mm + __logf(__expf(a0 - mm) + __expf(a1 - mm));
      const float f = -EPS_V * lse + eps_log_mu;
      F[k] = f;
      xmax0 = fmaxf(xmax0, (f - C0[k]) * inv_eps);
      xmax1 = fmaxf(xmax1, (f - C1[k]) * inv_eps);
    }
    // workgroup max reduction (for stable column logsumexp)
    for (int off = 16; off; off >>= 1) {
      xmax0 = fmaxf(xmax0, __shfl_xor(xmax0, off, 32));
      xmax1 = fmaxf(xmax1, __shfl_xor(xmax1, off, 32));
    }
    if (lane == 0) { redMax0[wid] = xmax0; redMax1[wid] = xmax1; }
    __syncthreads();
    float M0 = redMax0[0], M1 = redMax1[0];
    for (int w = 1; w < NWAVES; ++w) {
      M0 = fmaxf(M0, redMax0[w]);
      M1 = fmaxf(M1, redMax1[w]);
    }

    // per-lane partial sums of exp
    float p0 = 0.0f, p1 = 0.0f;
    for (int k = 0; k < EPT; ++k) {
      p0 += __expf((F[k] - C0[k]) * inv_eps - M0);
      p1 += __expf((F[k] - C1[k]) * inv_eps - M1);
    }

    // WMMA lane reduction: B layout (4x16 f32) per lane = {p0,p1};
    // D(0,n)=p0_n+p0_{n+16} in d[0] lanes 0..15, D(1,n)=p1_n+p1_{n+16} in
    // d[1]; rows 2..15 are zero, so a full 32-lane xor-sum finishes each.
    v2f bmat;
    bmat.x = p0;
    bmat.y = p1;
    v8f cacc = {};
    v8f dacc = __builtin_amdgcn_wmma_f32_16x16x4_f32(
        false, amat, false, bmat, (short)0, cacc, false, false);
    float s0 = dacc[0], s1 = dacc[1];
    for (int off = 16; off; off >>= 1) {
      s0 += __shfl_xor(s0, off, 32);
      s1 += __shfl_xor(s1, off, 32);
    }
    if (lane == 0) { redSum0[wid] = s0; redSum1[wid] = s1; }
    __syncthreads();
    float S0 = 0.0f, S1 = 0.0f;
    for (int w = 0; w < NWAVES; ++w) {
      S0 += redSum0[w];
      S1 += redSum1[w];
    }
    // g-update
    g0 = -EPS_V * (M0 + __logf(S0)) + eps_log_nu0;
    g1 = -EPS_V * (M1 + __logf(S1)) + eps_log_nu1;
  }

  // ---------------- Output: Gamma[...,0] * N ----------------
  for (int k = 0; k < EPT; ++k) {
    const int i = tid + k * TPB;
    const float gamma0 = __expf((F[k] + g0 - C0[k]) * inv_eps);
    out[(size_t)b * N_ELEM + i] = gamma0 * (float)N_ELEM;
  }
}

extern "C" void kernel_launch(void* const* d_in, const int* in_sizes, int n_in,
                              void* d_out, int out_size, void* d_ws, size_t ws_size,
                              hipStream_t stream) {
  const float* T   = (const float*)d_in[0];   // [B, N, 2,2,2] f32
  const int* state = (const int*)d_in[1];     // [B, N] i32
  float* out       = (float*)d_out;           // [B, N] f32

  const int Bn = in_sizes[1] / N_ELEM;        // 128
  rmab_sinkhorn_kernel<<<dim3(Bn), dim3(TPB), 0, stream>>>(T, state, out);
}